// TextProposal_43430709297349
// MI455X (gfx1250) — compile-verified
//
#include <hip/hip_runtime.h>
#include <stdint.h>

// Problem constants (match reference)
#define BB 16
#define NN 131072          // 2^17
#define KK 500
#define NEGF (-1e30f)
#define IOU_THR 0.3f
#define SCORE_THR 0.7f

typedef _Float16 h8 __attribute__((ext_vector_type(8)));   // 16B LDS vector

// ---------------------------------------------------------------------------
// Phase 1: gather + softmax + box regression.  Pure bandwidth, coalesced.
// ---------------------------------------------------------------------------
__global__ __launch_bounds__(256) void prep_kernel(
    const float* __restrict__ deltas, const float* __restrict__ logits,
    const float* __restrict__ anchors, const int* __restrict__ vai,
    float* __restrict__ boxes, float* __restrict__ active)
{
  long long gid = (long long)blockIdx.x * 256 + threadIdx.x;   // b*N + i
  int b = (int)(gid >> 17);
  int idx = vai[gid];
  long long gsrc = ((long long)b << 17) + idx;                 // gathered index
  float dy = deltas[gsrc * 2 + 0] * 0.1f;
  float dh = deltas[gsrc * 2 + 1] * 0.2f;
  float l0 = logits[gsrc * 2 + 0];
  float l1 = logits[gsrc * 2 + 1];
  float fg = 1.0f / (1.0f + __expf(l0 - l1));                  // softmax class-1
  float4 a = ((const float4*)anchors)[gid];                    // y1,x1,y2,x2
  float h  = a.z - a.x;
  float cy = (a.z + a.x) * 0.5f + dy * h;
  h *= __expf(dh);
  ((float4*)boxes)[gid] = make_float4(cy - 0.5f * h, a.y, cy + 0.5f * h, a.w);
  active[gid] = (fg >= SCORE_THR) ? fg : NEGF;
}

// ---------------------------------------------------------------------------
// CDNA5 async global->LDS copy (ASYNCcnt path) + split wait counters.
// ---------------------------------------------------------------------------
__device__ __forceinline__ void async_lds_b128(void* lds, const void* g) {
  unsigned loff = (unsigned)(size_t)lds;            // LDS byte offset (low 32b)
  unsigned long long ga = (unsigned long long)g;    // 64-bit global address
  asm volatile("global_load_async_to_lds_b128 %0, %1, off"
               :: "v"(loff), "v"(ga) : "memory");
}
__device__ __forceinline__ void wait_async0() {
  asm volatile("s_wait_asynccnt 0x0" ::: "memory");
}
__device__ __forceinline__ void wait_async1() {
  asm volatile("s_wait_asynccnt 0x1" ::: "memory");
}
__device__ __forceinline__ void wait_loads0() {
  asm volatile("s_wait_loadcnt 0x0" ::: "memory");
}

__device__ __forceinline__ void amax2(float& v, int& i, float v2, int i2) {
  // argmax with lowest-index tie-break (matches jnp.argmax); visitation-order
  // independent because ties always resolve by explicit index compare.
  if (v2 > v || (v2 == v && i2 < i)) { v = v2; i = i2; }
}

// ---------------------------------------------------------------------------
// Phase 2: ordered in-place compaction (score >= thr, ~27% survive) followed
// by greedy NMS over the compacted set.  One workgroup (32 wave32 waves) per
// image.  Compacted fp16 scores live in LDS for all 500 iterations (worst
// case 256 KB <= CDNA5's 320 KB WGP LDS); the argmax scan uses b128 LDS
// vector loads (8 halves/op) to hit LDS bandwidth instead of issue rate;
// compacted boxes stream from L2 via double-buffered async global->LDS
// copies overlapped with the IoU math.
// ---------------------------------------------------------------------------
__global__ __launch_bounds__(1024) void nms_kernel(
    float* __restrict__ boxes,     // [B,N,4] in, compacted in place
    float* __restrict__ active,    // [B,N]   in, compacted scores in place
    int*   __restrict__ cidx,      // [B,N]   compacted -> original index
    int*   __restrict__ sel, float* __restrict__ valid,
    float* __restrict__ selsc, float4* __restrict__ selbox)
{
  const int b = blockIdx.x;
  const int t = threadIdx.x;
  float4* boxb = (float4*)(boxes + (size_t)b * NN * 4);
  float*  actb = active + (size_t)b * NN;
  int*    cidb = cidx + (size_t)b * NN;

  extern __shared__ _Float16 s_act[];          // up to NN halves = 256 KB
  __shared__ unsigned s_off[32];
  __shared__ unsigned s_cnt;
  __shared__ float    s_redv[32];
  __shared__ int      s_redi[32];
  __shared__ float    s_selv;
  __shared__ float4   s_selbox;
  __shared__ float4   s_stage[2][1024];        // 32 KB async double buffer

  // ---- ordered, deterministic, in-place stream compaction ----------------
  if (t == 0) s_cnt = 0;
  __syncthreads();
  for (int k = 0; k < 128; ++k) {
    int j = (k << 10) + t;
    float v  = actb[j];
    float4 bx = boxb[j];
    wait_loads0();            // reads complete before in-place writes below
    bool f = v > NEGF;
    unsigned m32  = (unsigned)__ballot(f);     // wave32: low 32 bits
    unsigned lane = t & 31, w = t >> 5;
    unsigned pre  = __popc(m32 & ((1u << lane) - 1u));
    if (lane == 0) s_off[w] = __popc(m32);
    __syncthreads();
    if (t == 0) {
      unsigned acc = s_cnt;
      for (int i = 0; i < 32; ++i) { unsigned c = s_off[i]; s_off[i] = acc; acc += c; }
      s_cnt = acc;
    }
    __syncthreads();
    if (f) {
      unsigned pos = s_off[w] + pre;           // pos < (k+1)*1024: safe in place
      s_act[pos] = (_Float16)v;
      actb[pos]  = v;                          // exact f32 score, compacted
      boxb[pos]  = bx;                         // compacted boxes, coalesced
      cidb[pos]  = j;                          // original index
    }
    __syncthreads();
  }
  const int M  = (int)s_cnt;                   // survivors (data dependent)
  const int kM = (M + 1023) >> 10;             // suppression tiles (1024 elems)
  const int kS = M >> 13;                      // full vector-scan tiles (8192)

  // ---- greedy NMS over the compacted set ----------------------------------
  for (int it = 0; it < KK; ++it) {
    float bv = -__builtin_huge_valf();
    int   bp = 0x7fffffff;
    // vectorized argmax: 8 halves per ds_load_b128
    for (int k = 0; k < kS; ++k) {
      int base = (k << 13) + (t << 3);
      h8 hv = *(const h8*)(s_act + base);
      #pragma unroll
      for (int e = 0; e < 8; ++e) amax2(bv, bp, (float)hv[e], base + e);
    }
    // guarded scalar tail (< 8192 elements)
    for (int p = (kS << 13) + t; p < M; p += 1024)
      amax2(bv, bp, (float)s_act[p], p);
    // wave32 shuffle tree
    for (int off = 16; off; off >>= 1) {
      float ov = __shfl_xor(bv, off, 32);
      int   oi = __shfl_xor(bp, off, 32);
      amax2(bv, bp, ov, oi);
    }
    if ((t & 31) == 0) { s_redv[t >> 5] = bv; s_redi[t >> 5] = bp; }
    __syncthreads();
    if (t < 32) {                              // 32 waves exactly
      bv = s_redv[t]; bp = s_redi[t];
      for (int off = 16; off; off >>= 1) {
        float ov = __shfl_xor(bv, off, 32);
        int   oi = __shfl_xor(bp, off, 32);
        amax2(bv, bp, ov, oi);
      }
      if (t == 0) {
        s_selv = bv;
        bool ok = bv > NEGF;
        sel[b * KK + it]   = ok ? cidb[bp] : 0;
        valid[b * KK + it] = ok ? 1.0f : 0.0f;
        selsc[b * KK + it] = ok ? actb[bp] : 0.0f;
        if (ok) {
          // broadcast selected box via one-lane async copy into LDS
          async_lds_b128(&s_selbox, (const void*)(boxb + bp));
          wait_async0();
          selbox[b * KK + it] = s_selbox;      // persist for output phase
        } else {
          selbox[b * KK + it] = make_float4(0.f, 0.f, 0.f, 0.f);
        }
      }
    }
    __syncthreads();

    bool ok = s_selv > NEGF;                   // uniform across block
    if (ok) {
      float4 sb = s_selbox;
      float areaA = (sb.z - sb.x) * (sb.w - sb.y);
      int buf = 0;
      int pc0 = t < M ? t : M - 1;
      async_lds_b128(&s_stage[0][t], (const void*)(boxb + pc0));
      for (int k = 0; k < kM; ++k) {
        if (k + 1 < kM) {
          int pn  = ((k + 1) << 10) + t;
          int pcn = pn < M ? pn : M - 1;       // clamp => uniform issue pattern
          async_lds_b128(&s_stage[buf ^ 1][t], (const void*)(boxb + pcn));
          wait_async1();                       // tile k complete (in-order)
        } else {
          wait_async0();
        }
        float4 q = s_stage[buf][t];
        buf ^= 1;
        int p = (k << 10) + t;
        float yy1 = fmaxf(sb.x, q.x), xx1 = fmaxf(sb.y, q.y);
        float yy2 = fminf(sb.z, q.z), xx2 = fminf(sb.w, q.w);
        float inter = fmaxf(yy2 - yy1, 0.f) * fmaxf(xx2 - xx1, 0.f);
        float a2  = (q.z - q.x) * (q.w - q.y);
        float iou = inter / (areaA + a2 - inter + 1e-8f);
        if (p < M && iou > IOU_THR)
          s_act[p] = (_Float16)(-__builtin_huge_valf());
      }
    }
    __syncthreads();
  }
}

// ---------------------------------------------------------------------------
// Phase 3: format outputs ([B,500,5] | [B,500,2] | [B,500,3], flat).
// ---------------------------------------------------------------------------
__global__ __launch_bounds__(256) void out_kernel(
    const float4* __restrict__ selbox, const int* __restrict__ sel,
    const float* __restrict__ valid, const float* __restrict__ selsc,
    const float* __restrict__ logits, const int* __restrict__ vai,
    float* __restrict__ out)
{
  int gid = blockIdx.x * 256 + threadIdx.x;
  if (gid >= BB * KK) return;
  int b = gid / KK;
  int s = sel[gid];
  float m = valid[gid];
  float4 bx = selbox[gid];
  float* o0 = out + (size_t)gid * 5;                       // boxes + tag
  o0[0] = bx.x * m; o0[1] = bx.y * m; o0[2] = bx.z * m; o0[3] = bx.w * m; o0[4] = m;
  float* o1 = out + (size_t)BB * KK * 5 + (size_t)gid * 2; // scores + tag
  o1[0] = selsc[gid] * m; o1[1] = m;
  long long gsrc = ((long long)b << 17) + vai[((long long)b << 17) + s];
  float* o2 = out + (size_t)BB * KK * 7 + (size_t)gid * 3; // logits + tag
  o2[0] = logits[gsrc * 2 + 0] * m;
  o2[1] = logits[gsrc * 2 + 1] * m;
  o2[2] = m;
}

// ---------------------------------------------------------------------------
// Workspace (floats): boxes[B*N*4] | active[B*N] | cidx[B*N] (int) |
// sel[B*K] (int) | valid[B*K] | selsc[B*K] | selbox[B*K*4]  ->  ~51 MB.
// Everything read is written earlier in the same launch (deterministic).
// ---------------------------------------------------------------------------
extern "C" void kernel_launch(void* const* d_in, const int* in_sizes, int n_in,
                              void* d_out, int out_size, void* d_ws, size_t ws_size,
                              hipStream_t stream) {
  const float* deltas  = (const float*)d_in[0];
  const float* logits  = (const float*)d_in[1];
  const float* anchors = (const float*)d_in[2];
  const int*   vai     = (const int*)d_in[3];
  float* out = (float*)d_out;

  float*  ws_boxes  = (float*)d_ws;
  float*  ws_active = ws_boxes + (size_t)BB * NN * 4;
  int*    ws_cidx   = (int*)(ws_active + (size_t)BB * NN);
  int*    ws_sel    = ws_cidx + (size_t)BB * NN;
  float*  ws_valid  = (float*)(ws_sel + BB * KK);
  float*  ws_selsc  = ws_valid + BB * KK;
  float4* ws_selbox = (float4*)(ws_selsc + BB * KK);

  prep_kernel<<<(BB * NN) / 256, 256, 0, stream>>>(
      deltas, logits, anchors, vai, ws_boxes, ws_active);

  // 256 KB dynamic LDS: worst-case compacted score state resident in WGP LDS.
  nms_kernel<<<BB, 1024, (size_t)NN * sizeof(_Float16), stream>>>(
      ws_boxes, ws_active, ws_cidx, ws_sel, ws_valid, ws_selsc, ws_selbox);

  out_kernel<<<(BB * KK + 255) / 256, 256, 0, stream>>>(
      ws_selbox, ws_sel, ws_valid, ws_selsc, logits, vai, out);
}